// LinearPretrainHead_57939108823229
// MI455X (gfx1250) — compile-verified
//
#include <hip/hip_runtime.h>
#include <hip/hip_bf16.h>

typedef __attribute__((ext_vector_type(2))) float v2f;
typedef __attribute__((ext_vector_type(4))) float v4f;
typedef __attribute__((ext_vector_type(8))) float v8f;

#define D_MODEL 512
#define SEQ_LEN 1440
#define BEFF    896
#define NB      64
#define C_IN    14
#define LDS_PITCH 516   // 512 + 4 pad -> conflict-free float2 column reads

// ---------------------------------------------------------------------------
// GEMM + exp epilogue:  E[b][l*P + c] = exp( dot(X[m,:], W[c,:]) + bias[c] )
// X : [BEFF*L, 512] row-major, W : [P, 512] row-major, E : [BEFF, 1440]
// Block = 8 waves; each wave computes one 16x16 tile via v_wmma_f32_16x16x4_f32
// ---------------------------------------------------------------------------
__global__ __launch_bounds__(256) void gemm_exp_kernel(
    const float* __restrict__ X, const float* __restrict__ W,
    const float* __restrict__ bias, float* __restrict__ E,
    int L, int P)
{
    __shared__ float ldsW[16 * LDS_PITCH];

    const int tid  = threadIdx.x;
    const int wave = tid >> 5;
    const int lane = tid & 31;
    const int c0   = blockIdx.y * 16;           // output-column (W-row) tile base

    // Stage W[c0..c0+15][0..511] into LDS (zero pad past P). 2048 float4 slots.
    #pragma unroll
    for (int i = 0; i < 8; ++i) {
        int idx  = tid + i * 256;
        int row  = idx >> 7;                    // /128
        int col4 = (idx & 127) << 2;            // *4
        v4f v = {0.f, 0.f, 0.f, 0.f};
        if (c0 + row < P)
            v = *(const v4f*)(W + (size_t)(c0 + row) * D_MODEL + col4);
        *(v4f*)(ldsW + row * LDS_PITCH + col4) = v;
    }
    __syncthreads();

    const int laneHalf = lane >> 4;             // 0: K=k,k+1  1: K=k+2,k+3
    const int laneMod  = lane & 15;
    const int m0 = blockIdx.x * 128 + wave * 16;        // tile row base
    // A fragment source: row m0+laneMod, K offset 2*laneHalf  (ISA 16x4 layout)
    const float* ax = X + (size_t)(m0 + laneMod) * D_MODEL + 2 * laneHalf;
    // B fragment source: W-row laneMod from LDS, same K offset
    const float* bx = ldsW + laneMod * LDS_PITCH + 2 * laneHalf;

    v8f acc = {0.f, 0.f, 0.f, 0.f, 0.f, 0.f, 0.f, 0.f};

    for (int kk = 0; kk < D_MODEL; kk += 32) {
        __builtin_prefetch(ax + kk + 256, 0, 1);   // global_prefetch_b8 on X stream
        #pragma unroll
        for (int k = kk; k < kk + 32; k += 4) {
            v2f a = *(const v2f*)(ax + k);
            v2f b = *(const v2f*)(bx + k);
            acc = __builtin_amdgcn_wmma_f32_16x16x4_f32(
                      false, a, false, b, (short)0, acc, false, false);
        }
    }

    // D layout: VGPR r, lanes 0-15 -> M=r, lanes 16-31 -> M=r+8; N = lane&15
    const int col = c0 + laneMod;
    if (col < P) {
        const float bv = bias[col];
        #pragma unroll
        for (int r = 0; r < 8; ++r) {
            int m = m0 + r + 8 * laneHalf;
            int b = m / L;
            int l = m - b * L;
            int s = l * P + col;                 // sequence position
            E[(size_t)b * SEQ_LEN + s] = __expf(acc[r] + bv);
        }
    }
}

// ---------------------------------------------------------------------------
// combined = log( sum_e gates[be,e] * E_e[be,s] ),  ==0 -> float64 eps,
// laid out as out[b0][s][c] with be = b0*14 + c.
// ---------------------------------------------------------------------------
__global__ __launch_bounds__(256) void combine_kernel(
    const float* __restrict__ E,        // [4][BEFF][SEQ_LEN]
    const float* __restrict__ gates,    // [BEFF][4]
    float* __restrict__ out)            // [NB][SEQ_LEN][C_IN]
{
    int idx = blockIdx.x * blockDim.x + threadIdx.x;     // over NB*SEQ_LEN
    if (idx >= NB * SEQ_LEN) return;
    int b0 = idx / SEQ_LEN;
    int s  = idx - b0 * SEQ_LEN;

    #pragma unroll
    for (int c = 0; c < C_IN; ++c) {
        int be = b0 * C_IN + c;
        float acc = 0.f;
        #pragma unroll
        for (int e = 0; e < 4; ++e)
            acc += gates[be * 4 + e] *
                   E[((size_t)e * BEFF + be) * SEQ_LEN + s];
        acc = (acc == 0.f) ? 2.2204460492503131e-16f : acc;
        out[((size_t)b0 * SEQ_LEN + s) * C_IN + c] = __logf(acc);
    }
}

// ---------------------------------------------------------------------------
extern "C" void kernel_launch(void* const* d_in, const int* in_sizes, int n_in,
                              void* d_out, int out_size, void* d_ws, size_t ws_size,
                              hipStream_t stream) {
    (void)in_sizes; (void)n_in; (void)out_size; (void)ws_size;
    // d_in order (setup_inputs dict): x0,W0,b0, x1,W1,b1, x2,W2,b2, x3,W3,b3, gates
    float* E = (float*)d_ws;                    // [4][896][1440] f32 = 20.6 MB

    const int Ps[4] = {8, 16, 24, 48};
    for (int e = 0; e < 4; ++e) {
        const float* X    = (const float*)d_in[3 * e + 0];
        const float* W    = (const float*)d_in[3 * e + 1];
        const float* bias = (const float*)d_in[3 * e + 2];
        const int P = Ps[e];
        const int L = SEQ_LEN / P;              // exact for all scales
        const int Mtot = BEFF * L;              // divisible by 128 for all scales
        dim3 grid(Mtot / 128, (P + 15) / 16);
        gemm_exp_kernel<<<grid, 256, 0, stream>>>(
            X, W, bias, E + (size_t)e * BEFF * SEQ_LEN, L, P);
    }

    const float* gates = (const float*)d_in[12];
    int n = NB * SEQ_LEN;
    combine_kernel<<<(n + 255) / 256, 256, 0, stream>>>(E, gates, (float*)d_out);
}